// GDCN_55748675502368
// MI455X (gfx1250) — compile-verified
//
#include <hip/hip_runtime.h>
#include <math.h>

// ---------------------------------------------------------------------------
// 2-layer GAT + projection for MI455X (gfx1250, wave32).
//   * dense GEMMs: V_WMMA_F32_16X16X4_F32, W panel staged in LDS, 8 waves per
//     workgroup each computing a 16-row x K stripe (A fragment reused across
//     K/16 column tiles per WMMA step)
//   * edge softmax: u32-monotonic atomicMax + f32 atomicAdd segment ops
//   * weighted scatter-add: one wave per edge
// ---------------------------------------------------------------------------

#define GN      10000      // nodes  (625 * 16 exactly)
#define GE      640000     // edges before self loops
#define GETOT   (GE + GN)  // edges after self loops
#define D_IN    128
#define H1      128
#define H2      64
#define EMB     64
#define NEG_SLOPE 0.2f
#define ROW_TILES ((GN + 15) / 16)     // 625

typedef float v2f __attribute__((ext_vector_type(2)));
typedef float v8f __attribute__((ext_vector_type(8)));

// ---- monotonic float<->uint encoding for atomic max over signed floats ----
__device__ __forceinline__ unsigned f32_enc(float f) {
    unsigned u = __float_as_uint(f);
    return (u & 0x80000000u) ? ~u : (u | 0x80000000u);
}
__device__ __forceinline__ float f32_dec(unsigned e) {
    unsigned u = (e & 0x80000000u) ? (e ^ 0x80000000u) : ~e;
    return __uint_as_float(u);
}
#define ENC_NEG_INF 0x007FFFFFu   // f32_enc(-inf)

// ---------------------------------------------------------------------------
// Fill kernels (scratch re-initialized every call: graph-replay safe)
// ---------------------------------------------------------------------------
__global__ void fill_f32(float* __restrict__ p, float v, int n) {
    int i = blockIdx.x * blockDim.x + threadIdx.x;
    if (i < n) p[i] = v;
}
__global__ void fill_u32(unsigned* __restrict__ p, unsigned v, int n) {
    int i = blockIdx.x * blockDim.x + threadIdx.x;
    if (i < n) p[i] = v;
}

// ---------------------------------------------------------------------------
// WMMA fp32 GEMM: C[GN x KT] = A[GN x MT] * W[MT x KT] (+ bias)
//
// Workgroup = 256 threads = 8 waves. W (MT x KT, <= 64KB) is cooperatively
// staged in LDS once per block; each wave then computes a 16-row x KT stripe:
// per 4-wide k-step it loads one A fragment (2 VGPRs) and issues KT/16 WMMAs
// against B fragments pulled from LDS.
//
// Fragment layouts per ISA 7.12.2 (wave32):
//   A 16x4 : lanes 0-15 -> K=0(v0),1(v1); lanes 16-31 -> K=2(v0),3(v1)
//   B 4x16 : v0 = rows {0 | 2}, v1 = rows {1 | 3}, cols striped over l16
//   C/D    : VGPR g = row g (lanes 0-15), row g+8 (lanes 16-31)
// ---------------------------------------------------------------------------
template <int MT, int KT>
__global__ void gat_gemm_wmma(const float* __restrict__ A,
                              const float* __restrict__ W,
                              const float* __restrict__ bias,   // may be null
                              float* __restrict__ C) {
    __shared__ float sW[MT * KT];

    const int tid = threadIdx.x;

    // cooperative LDS fill of the weight panel (float4 vectorized)
    {
        const float4* __restrict__ W4 = (const float4*)W;
        float4* sW4 = (float4*)sW;
        const int total4 = (MT * KT) / 4;
        for (int i = tid; i < total4; i += 256) sW4[i] = W4[i];
    }
    __syncthreads();   // s_barrier_signal/s_barrier_wait (-1)

    const int wave = tid >> 5;
    const int lane = tid & 31;
    const int half = lane >> 4;     // 0|1
    const int l16  = lane & 15;
    const int rowTile = blockIdx.x * 8 + wave;

    if (rowTile < ROW_TILES) {      // uniform per-wave: EXEC all-ones inside
        const float* __restrict__ Arow =
            A + (size_t)(rowTile * 16 + l16) * MT + half * 2;

        v8f acc[KT / 16] = {};
        for (int kk = 0; kk < MT; kk += 4) {
            v2f a;
            a.x = Arow[kk + 0];
            a.y = Arow[kk + 1];
            __builtin_prefetch(Arow + kk + 16, 0, 0);   // global_prefetch_b8
            const float* __restrict__ sb0 = sW + (kk + half * 2) * KT + l16;
#pragma unroll
            for (int ct = 0; ct < KT / 16; ++ct) {
                v2f b;
                b.x = sb0[ct * 16];         // row kk+half*2
                b.y = sb0[ct * 16 + KT];    // row kk+half*2+1
                acc[ct] = __builtin_amdgcn_wmma_f32_16x16x4_f32(
                    /*neg_a=*/false, a, /*neg_b=*/false, b,
                    /*c_mod=*/(short)0, acc[ct],
                    /*reuse_a=*/false, /*reuse_b=*/false);
            }
        }

#pragma unroll
        for (int ct = 0; ct < KT / 16; ++ct) {
            const int col = ct * 16 + l16;
            const float bval = bias ? bias[col] : 0.0f;
            float* __restrict__ Crow =
                C + (size_t)(rowTile * 16 + half * 8) * KT + col;
#pragma unroll
            for (int g = 0; g < 8; ++g)
                Crow[(size_t)g * KT] = acc[ct][g] + bval;
        }
    }
}

// ---------------------------------------------------------------------------
// Per-node attention scalars: as[n] = h[n].a_src, ad[n] = h[n].a_dst
// ---------------------------------------------------------------------------
__global__ void node_dots(const float* __restrict__ h,
                          const float* __restrict__ a_src,
                          const float* __restrict__ a_dst,
                          float* __restrict__ out_s,
                          float* __restrict__ out_d,
                          int D) {
    const int lane = threadIdx.x & 31;
    const int node = blockIdx.x * (blockDim.x >> 5) + (threadIdx.x >> 5);
    if (node >= GN) return;
    float sa = 0.f, sd = 0.f;
    const float* __restrict__ hr = h + (size_t)node * D;
    for (int i = lane; i < D; i += 32) {
        const float hv = hr[i];
        sa += hv * a_src[i];
        sd += hv * a_dst[i];
    }
    for (int off = 16; off; off >>= 1) {
        sa += __shfl_xor(sa, off);
        sd += __shfl_xor(sd, off);
    }
    if (lane == 0) { out_s[node] = sa; out_d[node] = sd; }
}

__device__ __forceinline__ void edge_nodes(const long long* __restrict__ ei,
                                           int e, int& s, int& d) {
    if (e < GE) { s = (int)ei[e]; d = (int)ei[GE + e]; }
    else        { s = d = e - GE; }   // appended self-loop
}

__device__ __forceinline__ float edge_logit(const float* __restrict__ as_,
                                            const float* __restrict__ ad_,
                                            int s, int d) {
    const float v = as_[s] + ad_[d];
    return v > 0.f ? v : NEG_SLOPE * v;
}

// pass 1: segment max over destinations (monotonic-uint atomicMax)
__global__ void edge_max(const long long* __restrict__ ei,
                         const float* __restrict__ as_,
                         const float* __restrict__ ad_,
                         unsigned* __restrict__ m_enc) {
    const int e = blockIdx.x * blockDim.x + threadIdx.x;
    if (e >= GETOT) return;
    int s, d; edge_nodes(ei, e, s, d);
    atomicMax(&m_enc[d], f32_enc(edge_logit(as_, ad_, s, d)));
}

// pass 2: exp(logit - max) and segment sum
__global__ void edge_exp(const long long* __restrict__ ei,
                         const float* __restrict__ as_,
                         const float* __restrict__ ad_,
                         const unsigned* __restrict__ m_enc,
                         float* __restrict__ z,
                         float* __restrict__ ew) {
    const int e = blockIdx.x * blockDim.x + threadIdx.x;
    if (e >= GETOT) return;
    int s, d; edge_nodes(ei, e, s, d);
    const float w = __expf(edge_logit(as_, ad_, s, d) - f32_dec(m_enc[d]));
    ew[e] = w;
    atomicAdd(&z[d], w);
}

// pass 3: weighted scatter-add: agg[d] += (ew[e]/z[d]) * h[s]; wave per edge
__global__ void edge_agg(const long long* __restrict__ ei,
                         const float* __restrict__ h,
                         const float* __restrict__ ew,
                         const float* __restrict__ z,
                         float* __restrict__ agg,
                         int D) {
    const int lane = threadIdx.x & 31;
    const int e = blockIdx.x * (blockDim.x >> 5) + (threadIdx.x >> 5);
    if (e >= GETOT) return;
    int s, d; edge_nodes(ei, e, s, d);
    const float w = ew[e] / z[d];
    const float* __restrict__ hs = h + (size_t)s * D;
    float* __restrict__ ad2 = agg + (size_t)d * D;
    for (int i = lane; i < D; i += 32)
        atomicAdd(&ad2[i], w * hs[i]);
}

// epilogue: x = relu(agg + b)  (in place)
__global__ void bias_relu(float* __restrict__ p, const float* __restrict__ b,
                          int D, int n) {
    const int i = blockIdx.x * blockDim.x + threadIdx.x;
    if (i >= n) return;
    const float v = p[i] + b[i % D];
    p[i] = v > 0.f ? v : 0.f;
}

// ---------------------------------------------------------------------------
extern "C" void kernel_launch(void* const* d_in, const int* in_sizes, int n_in,
                              void* d_out, int out_size, void* d_ws, size_t ws_size,
                              hipStream_t stream) {
    (void)in_sizes; (void)n_in; (void)out_size; (void)ws_size;

    const float*     x     = (const float*)d_in[0];
    const long long* ei    = (const long long*)d_in[1];   // int64 [2, E]
    const float*     W1    = (const float*)d_in[2];
    const float*     asrc1 = (const float*)d_in[3];
    const float*     adst1 = (const float*)d_in[4];
    const float*     b1    = (const float*)d_in[5];
    const float*     W2    = (const float*)d_in[6];
    const float*     asrc2 = (const float*)d_in[7];
    const float*     adst2 = (const float*)d_in[8];
    const float*     b2    = (const float*)d_in[9];
    const float*     Wp    = (const float*)d_in[10];
    const float*     bp    = (const float*)d_in[11];
    float* out = (float*)d_out;

    // ---- workspace partition (floats) ----
    float* ws = (float*)d_ws;
    float*    h1    = ws;                 ws += (size_t)GN * H1;
    float*    x1    = ws;                 ws += (size_t)GN * H1;   // agg1 -> relu'd
    float*    h2    = ws;                 ws += (size_t)GN * H2;
    float*    x2    = ws;                 ws += (size_t)GN * H2;   // agg2 -> relu'd
    float*    asb   = ws;                 ws += GN;
    float*    adb   = ws;                 ws += GN;
    unsigned* m_enc = (unsigned*)ws;      ws += GN;
    float*    z     = ws;                 ws += GN;
    float*    ewb   = ws;                 ws += GETOT;

    const int T = 256;
    const int gemmBlocks  = (ROW_TILES + 7) / 8;               // 79
    const int edgeBlocks  = (GETOT + T - 1) / T;               // thread/edge
    const int edgeWaveBlk = (GETOT + (T / 32) - 1) / (T / 32); // wave/edge
    const int nodeWaveBlk = (GN + (T / 32) - 1) / (T / 32);

    // ================= layer 1 =================
    gat_gemm_wmma<D_IN, H1><<<gemmBlocks, T, 0, stream>>>(x, W1, nullptr, h1);
    node_dots<<<nodeWaveBlk, T, 0, stream>>>(h1, asrc1, adst1, asb, adb, H1);
    fill_u32<<<(GN + T - 1) / T, T, 0, stream>>>(m_enc, ENC_NEG_INF, GN);
    fill_f32<<<(GN + T - 1) / T, T, 0, stream>>>(z, 0.f, GN);
    fill_f32<<<((GN * H1) + T - 1) / T, T, 0, stream>>>(x1, 0.f, GN * H1);
    edge_max<<<edgeBlocks, T, 0, stream>>>(ei, asb, adb, m_enc);
    edge_exp<<<edgeBlocks, T, 0, stream>>>(ei, asb, adb, m_enc, z, ewb);
    edge_agg<<<edgeWaveBlk, T, 0, stream>>>(ei, h1, ewb, z, x1, H1);
    bias_relu<<<((GN * H1) + T - 1) / T, T, 0, stream>>>(x1, b1, H1, GN * H1);

    // ================= layer 2 =================
    gat_gemm_wmma<H1, H2><<<gemmBlocks, T, 0, stream>>>(x1, W2, nullptr, h2);
    node_dots<<<nodeWaveBlk, T, 0, stream>>>(h2, asrc2, adst2, asb, adb, H2);
    fill_u32<<<(GN + T - 1) / T, T, 0, stream>>>(m_enc, ENC_NEG_INF, GN);
    fill_f32<<<(GN + T - 1) / T, T, 0, stream>>>(z, 0.f, GN);
    fill_f32<<<((GN * H2) + T - 1) / T, T, 0, stream>>>(x2, 0.f, GN * H2);
    edge_max<<<edgeBlocks, T, 0, stream>>>(ei, asb, adb, m_enc);
    edge_exp<<<edgeBlocks, T, 0, stream>>>(ei, asb, adb, m_enc, z, ewb);
    edge_agg<<<edgeWaveBlk, T, 0, stream>>>(ei, h2, ewb, z, x2, H2);
    bias_relu<<<((GN * H2) + T - 1) / T, T, 0, stream>>>(x2, b2, H2, GN * H2);

    // ================= projection =================
    gat_gemm_wmma<H2, EMB><<<gemmBlocks, T, 0, stream>>>(x2, Wp, bp, out);
}